// MecanumSystemModel_56607668961318
// MI455X (gfx1250) — compile-verified
//
#include <hip/hip_runtime.h>
#include <math.h>

// ---------------------------------------------------------------------------
// Mecanum system model — pure HBM-streaming kernel (memory-bound: 60 B/elem,
// ~120 FLOP/elem -> bound by 23.3 TB/s HBM, floor ~5.2 us for B=2M).
// WMMA intentionally NOT used: the only matrix ops are 3x4 compile-time
// constants + a data-dependent 2x2 rotation; smallest WMMA tile (16x16xK)
// would require cross-lane packing costing more than the ~12 FMAs replaced.
// ---------------------------------------------------------------------------

namespace mk {
constexpr double G_    = 13.7;
constexpr double L_X   = 0.129907;
constexpr double L_Y   = 0.095724;
constexpr double R_    = 3.0 * 0.0254;              // 0.0762
constexpr double MASS_ = 12.0;
constexpr double MOI_  = MASS_ * (12.0 * 0.0254) * (12.0 * 0.0254) / 6.0;
constexpr double K_    = L_X + L_Y;                 // 0.225631

// wheel velocity map (rows of _local_to_wheel / R): wv = (lx ± ly ± K*w)/R
constexpr float INV_R   = (float)(1.0 / R_);
constexpr float K_INV_R = (float)(K_ / R_);

// torque = T_D*md - T_V*wv - T_F*softsign(wv)   (the outer *G folded in)
constexpr float T_D = (float)(G_ * 0.193);
constexpr float T_V = (float)(G_ * G_ * 0.000304);
constexpr float T_F = (float)(G_ * 0.00317);
constexpr float EPS = 0.01f;

// WHEEL_TORQUE_TO_LOCAL_ACCEL = pinv(local_to_wheel)/R/inertia.
// M^T M = diag(4,4,4K^2)  =>  rows are:
//   row0 = 1/(4 R MASS) * (1, 1, 1, 1)
//   row1 =       1/(4 R MASS) * (-1, 1, 1, -1)
//   row2 =       1/(4 K R MOI) * (-1, 1, -1, 1)
constexpr float ACC_XY = (float)(1.0 / (4.0 * R_ * MASS_));
constexpr float ACC_W  = (float)(1.0 / (4.0 * K_ * R_ * MOI_));
} // namespace mk

typedef float v2f __attribute__((ext_vector_type(2)));

__global__ __launch_bounds__(256) void mecanum_model_kernel(
    const float* __restrict__ state,   // (B,6): x y theta vx vy w
    const float* __restrict__ ctrl,    // (B,3)
    float* __restrict__ out,           // (B,6): vx vy w ax ay aw
    int B)
{
    const int i = blockIdx.x * blockDim.x + threadIdx.x;
    if (i >= B) return;

    // --- streaming loads (non-temporal: read-once data) -------------------
    // state[0..1] (x,y) are unused by the model -> never touch those regs.
    const float* sp = state + (size_t)i * 6;
    v2f sA = __builtin_nontemporal_load((const v2f*)(sp + 2)); // {theta, vx}
    v2f sB = __builtin_nontemporal_load((const v2f*)(sp + 4)); // {vy,    w }
    const float* up = ctrl + (size_t)i * 3;
    float u0 = __builtin_nontemporal_load(up + 0);
    float u1 = __builtin_nontemporal_load(up + 1);
    float u2 = __builtin_nontemporal_load(up + 2);

    const float theta = sA.x, vx = sA.y, vy = sB.x, wz = sB.y;

    // --- rotate world velocity into body frame (rot by -theta) -----------
    float s, c;
    __sincosf(theta, &s, &c);
    const float lx =  c * vx + s * vy;
    const float ly = -s * vx + c * vy;

    // --- wheel velocities --------------------------------------------------
    const float A = mk::INV_R * lx;
    const float Bv = mk::INV_R * ly;
    const float C = mk::K_INV_R * wz;
    const float wv0 = A - Bv - C;
    const float wv1 = A + Bv + C;
    const float wv2 = A + Bv - C;
    const float wv3 = A - Bv + C;

    // --- motor duties ------------------------------------------------------
    const float md0 = u0 - u1 - u2;
    const float md1 = u0 + u1 + u2;
    const float md2 = u0 + u1 - u2;
    const float md3 = u0 - u1 + u2;

    // --- wheel torques (with softsign friction) ---------------------------
    auto torque = [](float md, float wv) {
        const float ss = wv * rsqrtf(wv * wv + mk::EPS);
        return mk::T_D * md - mk::T_V * wv - mk::T_F * ss;
    };
    const float t0 = torque(md0, wv0);
    const float t1 = torque(md1, wv1);
    const float t2 = torque(md2, wv2);
    const float t3 = torque(md3, wv3);

    // --- body-frame acceleration (closed-form pinv rows) ------------------
    const float lax = mk::ACC_XY * ( t0 + t1 + t2 + t3);
    const float lay = mk::ACC_XY * (-t0 + t1 + t2 - t3);
    const float law = mk::ACC_W  * (-t0 + t1 - t2 + t3);

    // --- rotate back to world frame (rot by +theta) -----------------------
    const float ax = c * lax - s * lay;
    const float ay = s * lax + c * lay;

    // --- streaming stores (non-temporal: write-once data) -----------------
    float* op = out + (size_t)i * 6;
    v2f o0 = { vx, vy };
    v2f o1 = { wz, ax };
    v2f o2 = { ay, law };
    __builtin_nontemporal_store(o0, (v2f*)(op + 0));
    __builtin_nontemporal_store(o1, (v2f*)(op + 2));
    __builtin_nontemporal_store(o2, (v2f*)(op + 4));
}

extern "C" void kernel_launch(void* const* d_in, const int* in_sizes, int n_in,
                              void* d_out, int out_size, void* d_ws, size_t ws_size,
                              hipStream_t stream) {
    // setup_inputs order: t (1), state (B*6), control_duty (B*3)
    const float* state = (const float*)d_in[1];
    const float* ctrl  = (const float*)d_in[2];
    float* out = (float*)d_out;
    const int B = in_sizes[1] / 6;

    const int threads = 256;               // 8 wave32s per block
    const int blocks  = (B + threads - 1) / threads;
    mecanum_model_kernel<<<blocks, threads, 0, stream>>>(state, ctrl, out, B);
}